// SOM_86028194938904
// MI455X (gfx1250) — compile-verified
//
#include <hip/hip_runtime.h>

#define B_SZ   4096
#define D_SZ   512
#define M_SZ   16384
#define NBMU   16
#define EPSV   1e-6f

#define ROWS   32       // batch rows per block
#define MHALF  (M_SZ / 2)
#define NC     256      // columns per block per iteration
#define XPITCH 516      // A-tile LDS pitch (floats): conflict-free b64 frag reads
#define KPITCH 264      // key-buffer pitch; %64==8 -> conflict-free selection reads

typedef __attribute__((ext_vector_type(2))) float v2f;
typedef __attribute__((ext_vector_type(8))) float v8f;

// ---------------- Kernel 1: per-column key constants c[m] = sum(w^2) - 2*eps*sum(w)
__global__ void som_colstats(const float* __restrict__ w,
                             float* __restrict__ cbuf,
                             float* __restrict__ loss) {
    int m = blockIdx.x * blockDim.x + threadIdx.x;
    if (m < M_SZ) {
        float sq = 0.f, sm = 0.f;
        #pragma unroll 8
        for (int d = 0; d < D_SZ; ++d) {
            float v = w[(size_t)d * M_SZ + m];
            sq += v * v;
            sm += v;
        }
        cbuf[m] = sq - 2.0f * EPSV * sm;
    }
    if (blockIdx.x == 0 && threadIdx.x == 0) loss[0] = 0.0f;
}

// total-order "less" matching stable argsort (ties -> smaller index first)
__device__ __forceinline__ bool klt(float ka, int ia, float kb, int ib) {
    return (ka < kb) || (ka == kb && ia < ib);
}

__device__ __forceinline__ void top16_insert(float k, int m, float kk[16], int ii[16]) {
    if (klt(k, m, kk[15], ii[15])) {
        float ck = k; int ci = m;
        #pragma unroll
        for (int s = 0; s < 16; ++s) {           // unrolled bubble-insert, regs only
            bool sw = klt(ck, ci, kk[s], ii[s]);
            float tk = kk[s]; int ti = ii[s];
            kk[s] = sw ? ck : tk;  ii[s] = sw ? ci : ti;
            ck    = sw ? tk : ck;  ci    = sw ? ti : ti == ti ? ci : ci;
        }
    }
}

// ---------------- Kernel 2: fused f32-WMMA GEMM + streaming top-16 (per M-half)
__global__ __launch_bounds__(256)
void som_bmu_partial(const float* __restrict__ x, const float* __restrict__ w,
                     const float* __restrict__ cbuf,
                     float* __restrict__ part_key, int* __restrict__ part_idx) {
    __shared__ float s_x[ROWS * XPITCH];         // 64.5 KB; reused as merge buffer
    __shared__ float s_key[ROWS * KPITCH];       // 33 KB
    float* s_mkey = s_x;                         // 32*128 floats
    int*   s_midx = (int*)(s_x + ROWS * 128);    // 32*128 ints

    const int tid    = threadIdx.x;
    const int lane   = tid & 31;
    const int wv     = tid >> 5;           // 8 waves
    const int ln16   = lane & 15;
    const int klocal = 2 * (lane >> 4);    // 32-bit WMMA layout: half-wave K split
    const int rb     = blockIdx.x >> 1;
    const int mh     = blockIdx.x & 1;
    const int b0     = rb * ROWS;
    const int m0     = mh * MHALF;

    // ---- stage x tile [32 x 512] into LDS (coalesced b128) ----
    const float4* xg = (const float4*)(x + (size_t)b0 * D_SZ);
    #pragma unroll
    for (int i = 0; i < 16; ++i) {
        int idx = tid + 256 * i;           // 4096 float4s total
        int r = idx >> 7, c4 = idx & 127;
        float4 v = xg[r * 128 + c4];
        *(float4*)&s_x[r * XPITCH + 4 * c4] = v;
    }
    __syncthreads();

    const float* abase0 = &s_x[ln16 * XPITCH + klocal];
    const float* abase1 = abase0 + 16 * XPITCH;

    float kk[16]; int ii[16];
    #pragma unroll
    for (int s = 0; s < 16; ++s) { kk[s] = __builtin_inff(); ii[s] = 0x7fffffff; }

    const int row_sel = tid >> 3;          // 8 selection threads per batch row
    const int t_sel   = tid & 7;

    const v8f zero8 = {0.f, 0.f, 0.f, 0.f, 0.f, 0.f, 0.f, 0.f};

    for (int it = 0; it < MHALF / NC; ++it) {
        const int mbase = m0 + it * NC;

        // ---- GEMM: each wave -> 2 row-tiles x 2 col-tiles, K=512 via 16x16x4 f32
        v8f acc00 = zero8, acc01 = zero8, acc10 = zero8, acc11 = zero8;
        const float* wb0 = w + (size_t)klocal * M_SZ + (mbase + wv * 32 + ln16);
        const float* wb1 = wb0 + 16;

        #pragma unroll 4
        for (int k0 = 0; k0 < D_SZ; k0 += 4) {
            v2f a0 = *(const v2f*)(abase0 + k0);   // ds_load_b64
            v2f a1 = *(const v2f*)(abase1 + k0);
            v2f b0, b1;
            b0.x = wb0[(size_t)k0 * M_SZ];
            b0.y = wb0[(size_t)(k0 + 1) * M_SZ];
            b1.x = wb1[(size_t)k0 * M_SZ];
            b1.y = wb1[(size_t)(k0 + 1) * M_SZ];
            acc00 = __builtin_amdgcn_wmma_f32_16x16x4_f32(false, a0, false, b0, (short)0, acc00, false, false);
            acc10 = __builtin_amdgcn_wmma_f32_16x16x4_f32(false, a1, false, b0, (short)0, acc10, false, false);
            acc01 = __builtin_amdgcn_wmma_f32_16x16x4_f32(false, a0, false, b1, (short)0, acc01, false, false);
            acc11 = __builtin_amdgcn_wmma_f32_16x16x4_f32(false, a1, false, b1, (short)0, acc11, false, false);
        }

        // ---- write keys: key = c[m] - 2*xw (monotone in squared distance)
        {
            int col0 = wv * 32 + ln16;
            int col1 = col0 + 16;
            float cv0 = cbuf[mbase + col0];
            float cv1 = cbuf[mbase + col1];
            int rbase = 8 * (lane >> 4);
            #pragma unroll
            for (int i = 0; i < 8; ++i) {
                int r = i + rbase;
                s_key[r * KPITCH + col0]        = cv0 - 2.0f * acc00[i];
                s_key[(r + 16) * KPITCH + col0] = cv0 - 2.0f * acc10[i];
                s_key[r * KPITCH + col1]        = cv1 - 2.0f * acc01[i];
                s_key[(r + 16) * KPITCH + col1] = cv1 - 2.0f * acc11[i];
            }
        }
        __syncthreads();

        // ---- selection: each thread streams 32 candidates of its row ----
        #pragma unroll 4
        for (int jj = 0; jj < NC / 8; ++jj) {
            int col_l = t_sel + 8 * jj;
            float k = s_key[row_sel * KPITCH + col_l];
            int m = mbase + col_l;
            if (klt(k, m, kk[15], ii[15])) {
                float ck = k; int ci = m;
                #pragma unroll
                for (int s = 0; s < 16; ++s) {
                    bool sw = klt(ck, ci, kk[s], ii[s]);
                    float tk = kk[s]; int ti = ii[s];
                    kk[s] = sw ? ck : tk;  ii[s] = sw ? ci : ti;
                    ck    = sw ? tk : ck;  ci    = sw ? ti : ci;
                }
            }
        }
        __syncthreads();
    }

    // ---- dump per-thread lists into (dead) x-tile LDS ----
    #pragma unroll
    for (int s = 0; s < 16; ++s) {
        s_mkey[row_sel * 128 + t_sel * 16 + s] = kk[s];
        s_midx[row_sel * 128 + t_sel * 16 + s] = ii[s];
    }
    __syncthreads();

    // ---- per-row merge of 8 sorted lists -> partial top-16 to workspace ----
    if (tid < ROWS) {
        const int r = tid;
        #pragma unroll
        for (int s = 0; s < 16; ++s) { kk[s] = __builtin_inff(); ii[s] = 0x7fffffff; }
        for (int q = 0; q < 128; ++q) {
            float k = s_mkey[r * 128 + q];
            int m   = s_midx[r * 128 + q];
            if (klt(k, m, kk[15], ii[15])) {
                float ck = k; int ci = m;
                #pragma unroll
                for (int s = 0; s < 16; ++s) {
                    bool sw = klt(ck, ci, kk[s], ii[s]);
                    float tk = kk[s]; int ti = ii[s];
                    kk[s] = sw ? ck : tk;  ii[s] = sw ? ci : ti;
                    ck    = sw ? tk : ck;  ci    = sw ? ti : ci;
                }
            }
        }
        const size_t base = ((size_t)(b0 + r) * 2 + mh) * 16;
        #pragma unroll
        for (int s = 0; s < 16; ++s) {
            part_key[base + s] = kk[s];
            part_idx[base + s] = ii[s];
        }
    }
}

// ---------------- Kernel 3: merge the two M-half partials, gather locations
__global__ void som_merge(const float* __restrict__ part_key,
                          const int* __restrict__ part_idx,
                          const float* __restrict__ loc,
                          float* __restrict__ out) {
    const int b = blockIdx.x * blockDim.x + threadIdx.x;
    if (b >= B_SZ) return;

    float kk[16]; int ii[16];
    #pragma unroll
    for (int s = 0; s < 16; ++s) { kk[s] = __builtin_inff(); ii[s] = 0x7fffffff; }

    const size_t base = (size_t)b * 32;
    for (int q = 0; q < 32; ++q) {
        float k = part_key[base + q];
        int m   = part_idx[base + q];
        if (klt(k, m, kk[15], ii[15])) {
            float ck = k; int ci = m;
            #pragma unroll
            for (int s = 0; s < 16; ++s) {
                bool sw = klt(ck, ci, kk[s], ii[s]);
                float tk = kk[s]; int ti = ii[s];
                kk[s] = sw ? ck : tk;  ii[s] = sw ? ci : ti;
                ck    = sw ? tk : ck;  ci    = sw ? ti : ci;
            }
        }
    }

    const float2* lp = (const float2*)loc;
    #pragma unroll
    for (int s = 0; s < 16; ++s) {
        float2 L = lp[ii[s]];
        out[((size_t)s * B_SZ + b) * 2 + 0] = L.x;
        out[((size_t)s * B_SZ + b) * 2 + 1] = L.y;
    }
}

extern "C" void kernel_launch(void* const* d_in, const int* in_sizes, int n_in,
                              void* d_out, int out_size, void* d_ws, size_t ws_size,
                              hipStream_t stream) {
    const float* x   = (const float*)d_in[0];   // [B, D]
    const float* w   = (const float*)d_in[1];   // [D, M]
    const float* loc = (const float*)d_in[2];   // [M, 2]
    float* out  = (float*)d_out;                // [16, B, 2] ++ [loss]
    float* loss = out + (out_size - 1);

    // workspace carve: cbuf[M] | part_key[B*32] | part_idx[B*32]
    float* cbuf     = (float*)d_ws;
    float* part_key = cbuf + M_SZ;
    int*   part_idx = (int*)(part_key + (size_t)B_SZ * 32);

    som_colstats<<<M_SZ / 256, 256, 0, stream>>>(w, cbuf, loss);
    som_bmu_partial<<<(B_SZ / ROWS) * 2, 256, 0, stream>>>(x, w, cbuf, part_key, part_idx);
    som_merge<<<B_SZ / 256, 256, 0, stream>>>(part_key, part_idx, loc, out);
}